// SparseLinear_47072841564548
// MI455X (gfx1250) — compile-verified
//
#include <hip/hip_runtime.h>
#include <hip/hip_bf16.h>
#include <stdint.h>

#define WAVES_PER_BLOCK 8
#define BLOCK_THREADS   (WAVES_PER_BLOCK * 32)
#define N_FEATS         64
#define OUT_DIM         128

typedef __attribute__((ext_vector_type(4))) float v4f;

__global__ __launch_bounds__(BLOCK_THREADS)
void SparseLinear_gather_kernel(const int* __restrict__ indices,   // [B, 64] int32
                                const float* __restrict__ weight,  // [IN, 128] f32
                                const float* __restrict__ bias,    // [128] f32
                                float* __restrict__ out,           // [B, 128] f32
                                int batch)
{
    __shared__ int sidx[WAVES_PER_BLOCK * N_FEATS];

    const int lane = threadIdx.x & 31;
    const int wave = threadIdx.x >> 5;
    const int b    = blockIdx.x * WAVES_PER_BLOCK + wave;
    if (b >= batch) return;

    // ---- CDNA5 async copy: stage this sample's 64 i32 indices into LDS ----
    // Each lane moves 8 bytes (2 indices); one async-to-LDS op per wave.
    {
        uint64_t gaddr  = (uint64_t)(const void*)(indices + (size_t)b * N_FEATS + 2 * lane);
        unsigned ldsoff = (unsigned)(uint64_t)(void*)(&sidx[wave * N_FEATS + 2 * lane]);
        asm volatile("global_load_async_to_lds_b64 %0, %1, off"
                     :: "v"(ldsoff), "v"(gaddr) : "memory");
        asm volatile("s_wait_asynccnt 0" ::: "memory");
    }

    const int* slot = &sidx[wave * N_FEATS];
    const v4f* wrow = (const v4f*)weight;          // 32 x v4f per embedding row

    // bias for this lane's 4 output columns
    v4f acc = ((const v4f*)bias)[lane];

    // ---- gather-accumulate: one coalesced b128 load per lane per feature ----
    #pragma unroll 8
    for (int f = 0; f < N_FEATS; ++f) {
        // wave-uniform index -> SGPR, so the row base is scalar (GVS addressing)
        int idx = __builtin_amdgcn_readfirstlane(slot[f]);
        v4f r = wrow[(size_t)idx * (OUT_DIM / 4) + lane];
        acc += r;
    }

    // write-once output: non-temporal b128 store (keep L2 for the weight table)
    __builtin_nontemporal_store(acc, (v4f*)out + (size_t)b * (OUT_DIM / 4) + lane);
}

extern "C" void kernel_launch(void* const* d_in, const int* in_sizes, int n_in,
                              void* d_out, int out_size, void* d_ws, size_t ws_size,
                              hipStream_t stream) {
    const int*   indices = (const int*)d_in[0];
    const float* weight  = (const float*)d_in[1];
    const float* bias    = (const float*)d_in[2];
    float*       out     = (float*)d_out;

    const int batch = in_sizes[0] / N_FEATS;   // 16384
    const int grid  = (batch + WAVES_PER_BLOCK - 1) / WAVES_PER_BLOCK;

    SparseLinear_gather_kernel<<<grid, BLOCK_THREADS, 0, stream>>>(
        indices, weight, bias, out, batch);
}